// GraphAttentionLayer_78374563218167
// MI455X (gfx1250) — compile-verified
//
#include <hip/hip_runtime.h>
#include <hip/hip_bf16.h>
#include <math.h>

// Problem constants (from reference setup_inputs)
#define B_   2
#define N_   512
#define FIN  256
#define FOUT 128
#define LRELU_ALPHA 0.2f

typedef __attribute__((ext_vector_type(2))) float v2f;
typedef __attribute__((ext_vector_type(8))) float v8f;

// ---------------------------------------------------------------------------
// Kernel 1: Wh = x @ W   (rows = B*N over x, cols = FOUT)
// One wave32 per 16x16 output tile, K-loop of V_WMMA_F32_16X16X4_F32.
// A layout (f32 16x4): lane&15 = M; lanes>=16 hold K+2 half.
// B layout (f32 4x16): lane&15 = N; lanes>=16 hold K+2 half.
// C/D layout: VGPR r -> rows r (lanes 0-15) and r+8 (lanes 16-31).
// ---------------------------------------------------------------------------
__global__ __launch_bounds__(32) void wh_gemm(const float* __restrict__ x,
                                              const float* __restrict__ W,
                                              float* __restrict__ Wh) {
    const int lane = threadIdx.x;
    const int ln   = lane & 15;
    const int kh   = (lane >> 4) << 1;   // 0 or 2
    const int row0 = blockIdx.x * 16;    // over B*N
    const int col0 = blockIdx.y * 16;    // over FOUT
    const float* __restrict__ arow = x + (size_t)(row0 + ln) * FIN;

    v8f acc = {};
    for (int k0 = 0; k0 < FIN; k0 += 4) {
        v2f a, b;
        a.x = arow[k0 + kh];
        a.y = arow[k0 + kh + 1];
        b.x = W[(size_t)(k0 + kh) * FOUT + col0 + ln];
        b.y = W[(size_t)(k0 + kh + 1) * FOUT + col0 + ln];
        acc = __builtin_amdgcn_wmma_f32_16x16x4_f32(false, a, false, b,
                                                    (short)0, acc, false, false);
    }
    const int mh = (lane >> 4) << 3;     // 0 or 8
#pragma unroll
    for (int r = 0; r < 8; ++r) {
        Wh[(size_t)(row0 + r + mh) * FOUT + col0 + ln] = acc[r];
    }
}

// ---------------------------------------------------------------------------
// Kernel 2: zero the global sum-of-squares accumulator
// ---------------------------------------------------------------------------
__global__ void init_sumsq(float* sumsq) { *sumsq = 0.0f; }

// ---------------------------------------------------------------------------
// Kernel 3: e[b,i,j] = sum_o a[o] * leaky_relu(Wh[b,i,o] + Wh[b,j,o])
// 16x16 (i,j) tile per 256-thread block. Wh row tiles staged in LDS with
// padding (stride 132 -> bank index (4*tx + o) % 64, conflict-free).
// Block-reduce e^2 and atomically accumulate the Frobenius sum.
// ---------------------------------------------------------------------------
__global__ __launch_bounds__(256) void attn_scores(const float* __restrict__ Wh,
                                                   const float* __restrict__ afc,
                                                   float* __restrict__ e,
                                                   float* __restrict__ sumsq) {
    __shared__ float shi[16][FOUT + 4];
    __shared__ float shj[16][FOUT + 4];
    __shared__ float sa[FOUT];
    __shared__ float red[256];

    const int b   = blockIdx.z;
    const int i0  = blockIdx.y * 16;
    const int j0  = blockIdx.x * 16;
    const int tid = threadIdx.y * 16 + threadIdx.x;
    const float* __restrict__ WhB = Wh + (size_t)b * N_ * FOUT;

    for (int t = tid; t < 16 * FOUT; t += 256) {
        const int r = t >> 7;      // /128
        const int c = t & (FOUT - 1);
        shi[r][c] = WhB[(size_t)(i0 + r) * FOUT + c];
        shj[r][c] = WhB[(size_t)(j0 + r) * FOUT + c];
    }
    if (tid < FOUT) sa[tid] = afc[tid];
    __syncthreads();

    const int ti = threadIdx.y, tj = threadIdx.x;
    float acc = 0.0f;
#pragma unroll 4
    for (int o = 0; o < FOUT; ++o) {
        const float s = shi[ti][o] + shj[tj][o];
        const float l = (s > 0.0f) ? s : (LRELU_ALPHA * s);
        acc = fmaf(sa[o], l, acc);
    }
    e[((size_t)(b * N_ + i0 + ti)) * N_ + j0 + tj] = acc;

    red[tid] = acc * acc;
    __syncthreads();
    for (int s = 128; s > 0; s >>= 1) {
        if (tid < s) red[tid] += red[tid + s];
        __syncthreads();
    }
    if (tid == 0) atomicAdd(sumsq, red[0]);
}

// ---------------------------------------------------------------------------
// Kernel 4: per-row softmax with global-norm scaling and adj masking,
// written in-place over e (each block owns one (b,i) row of 512 entries).
// ---------------------------------------------------------------------------
__global__ __launch_bounds__(256) void softmax_rows(float* __restrict__ e,
                                                    const int* __restrict__ adj,
                                                    const float* __restrict__ sumsq) {
    __shared__ float red[256];
    const int b   = blockIdx.y;
    const int i   = blockIdx.x;
    const int tid = threadIdx.x;
    float* __restrict__ row = e + ((size_t)(b * N_ + i)) * N_;
    const int* __restrict__ arow = adj + ((size_t)(b * N_ + i)) * N_;

    const float scale = 1.0f / sqrtf(*sumsq);

    float v0 = (arow[tid]       == 0) ? -INFINITY : row[tid]       * scale;
    float v1 = (arow[tid + 256] == 0) ? -INFINITY : row[tid + 256] * scale;

    // row max
    red[tid] = fmaxf(v0, v1);
    __syncthreads();
    for (int s = 128; s > 0; s >>= 1) {
        if (tid < s) red[tid] = fmaxf(red[tid], red[tid + s]);
        __syncthreads();
    }
    const float m = red[0];
    __syncthreads();

    const float e0 = expf(v0 - m);
    const float e1 = expf(v1 - m);

    // row sum
    red[tid] = e0 + e1;
    __syncthreads();
    for (int s = 128; s > 0; s >>= 1) {
        if (tid < s) red[tid] += red[tid + s];
        __syncthreads();
    }
    const float inv = 1.0f / red[0];

    row[tid]       = e0 * inv;
    row[tid + 256] = e1 * inv;
}

// ---------------------------------------------------------------------------
// Kernel 5: h' = attention @ Wh, fused ELU at the store.
// Same wave-per-16x16-tile WMMA f32 structure as kernel 1, K = N_.
// ---------------------------------------------------------------------------
__global__ __launch_bounds__(32) void hprime_gemm(const float* __restrict__ att,
                                                  const float* __restrict__ Wh,
                                                  float* __restrict__ out) {
    const int lane = threadIdx.x;
    const int ln   = lane & 15;
    const int kh   = (lane >> 4) << 1;   // 0 or 2
    const int b    = blockIdx.z;
    const int row0 = blockIdx.x * 16;    // over N
    const int col0 = blockIdx.y * 16;    // over FOUT
    const float* __restrict__ A  = att + (size_t)b * N_ * N_;
    const float* __restrict__ Bm = Wh  + (size_t)b * N_ * FOUT;
    const float* __restrict__ arow = A + (size_t)(row0 + ln) * N_;

    v8f acc = {};
    for (int k0 = 0; k0 < N_; k0 += 4) {
        v2f a, bb;
        a.x  = arow[k0 + kh];
        a.y  = arow[k0 + kh + 1];
        bb.x = Bm[(size_t)(k0 + kh) * FOUT + col0 + ln];
        bb.y = Bm[(size_t)(k0 + kh + 1) * FOUT + col0 + ln];
        acc = __builtin_amdgcn_wmma_f32_16x16x4_f32(false, a, false, bb,
                                                    (short)0, acc, false, false);
    }
    const int mh = (lane >> 4) << 3;     // 0 or 8
    float* __restrict__ orow = out + (size_t)b * N_ * FOUT;
#pragma unroll
    for (int r = 0; r < 8; ++r) {
        const float v = acc[r];
        orow[(size_t)(row0 + r + mh) * FOUT + col0 + ln] =
            (v > 0.0f) ? v : expm1f(v);
    }
}

// ---------------------------------------------------------------------------
extern "C" void kernel_launch(void* const* d_in, const int* in_sizes, int n_in,
                              void* d_out, int out_size, void* d_ws, size_t ws_size,
                              hipStream_t stream) {
    (void)in_sizes; (void)n_in; (void)out_size; (void)ws_size;

    const float* x   = (const float*)d_in[0];   // [B,N,FIN]
    const int*   adj = (const int*)  d_in[1];   // [B,N,N]
    const float* W   = (const float*)d_in[2];   // [FIN,FOUT]
    const float* afc = (const float*)d_in[3];   // [FOUT]
    float*       out = (float*)d_out;           // [B,N,FOUT]

    char* ws = (char*)d_ws;
    float* Wh    = (float*)ws;                                           // B*N*FOUT
    float* e     = (float*)(ws + (size_t)B_ * N_ * FOUT * sizeof(float)); // B*N*N
    float* sumsq = (float*)(ws + ((size_t)B_ * N_ * FOUT +
                                  (size_t)B_ * N_ * N_) * sizeof(float)); // 1

    wh_gemm<<<dim3((B_ * N_) / 16, FOUT / 16, 1), 32, 0, stream>>>(x, W, Wh);
    init_sumsq<<<1, 1, 0, stream>>>(sumsq);
    attn_scores<<<dim3(N_ / 16, N_ / 16, B_), dim3(16, 16), 0, stream>>>(Wh, afc, e, sumsq);
    softmax_rows<<<dim3(N_, B_), 256, 0, stream>>>(e, adj, sumsq);
    hprime_gemm<<<dim3(N_ / 16, FOUT / 16, B_), 32, 0, stream>>>(e, Wh, out);
}